// DoubleDecoder_48163763257556
// MI455X (gfx1250) — compile-verified
//
#include <hip/hip_runtime.h>
#include <hip/hip_bf16.h>
#include <math.h>

// CDNA5 / gfx1250: wave32, WMMA f32 16x16x4, native v_tanh_f32 path.
typedef __attribute__((ext_vector_type(2))) float v2f;
typedef __attribute__((ext_vector_type(8))) float v8f;

#define WAVES_PER_BLOCK 2
#define LANES 32
#define M_PER_WAVE 32   // two 16-row WMMA M-tiles per wave (B-fragment reuse x2)
#define STRIDE 113      // odd stride (>=112) -> LDS bank-conflict-free rows

static __device__ __forceinline__ v8f wmma4(v2f a, v2f b, v8f c) {
  // (neg_a, A, neg_b, B, c_mod, C, reuse_a, reuse_b) -> v_wmma_f32_16x16x4_f32
  return __builtin_amdgcn_wmma_f32_16x16x4_f32(false, a, false, b, (short)0, c,
                                               false, false);
}

// Same-wave LDS write->read phase boundary (DS is in-order per wave; keep the
// compiler from reordering and make the dscnt dependency explicit).
static __device__ __forceinline__ void lds_fence() {
  asm volatile("s_wait_dscnt 0x0" ::: "memory");
}

// Native gfx1250 transcendental tanh (single TRANS op) if the toolchain
// exposes it; otherwise branch-free exp2/rcp formulation (5 VALU, no branches).
static __device__ __forceinline__ float fast_tanh(float x) {
#if __has_builtin(__builtin_amdgcn_tanhf)
  return __builtin_amdgcn_tanhf(x);
#else
  // tanh(x) = 1 - 2/(exp2(2*log2(e)*x) + 1); saturates correctly at +-inf.
  float e = __builtin_amdgcn_exp2f(x * 2.885390081777927f);
  return __builtin_fmaf(-2.f, __builtin_amdgcn_rcpf(e + 1.f), 1.f);
#endif
}

// Branchless padded load: unconditional global load from a cndmask'd address,
// then cndmask the value. No exec-mask save/restore, no branches.
static __device__ __forceinline__ float gload0(const float* __restrict__ p,
                                               bool valid, int idx) {
  float v = p[valid ? idx : 0];
  return valid ? v : 0.f;
}

// Accumulate two 16x16 C tiles (M-tiles 0 and 1) for output-column tile `nt`.
// A fragments (16x4 f32): lanes 0-15 hold K=k0,k0+1; lanes 16-31 K=k0+2,k0+3.
// B fragments (4x16 f32): mirrored; this lane's column is `col`.
template <int KI, int NI, int KPAD>
static __device__ __forceinline__ void tile_pair(
    const float* __restrict__ W, const float* __restrict__ hin, int col,
    int half, int l16, v8f& c0, v8f& c1) {
  c0 = (v8f){0.f, 0.f, 0.f, 0.f, 0.f, 0.f, 0.f, 0.f};
  c1 = (v8f){0.f, 0.f, 0.f, 0.f, 0.f, 0.f, 0.f, 0.f};
#pragma unroll
  for (int k0 = 0; k0 < KPAD; k0 += 4) {
    const int ka = k0 + half * 2;
    v2f b;  // shared by both M-tiles
    b.x = gload0(W, (ka + 0) < KI && col < NI, (ka + 0) * NI + col);
    b.y = gload0(W, (ka + 1) < KI && col < NI, (ka + 1) * NI + col);
    v2f a0, a1;
    a0.x = hin[l16 * STRIDE + ka];
    a0.y = hin[l16 * STRIDE + ka + 1];
    a1.x = hin[(16 + l16) * STRIDE + ka];
    a1.y = hin[(16 + l16) * STRIDE + ka + 1];
    c0 = wmma4(a0, b, c0);
    c1 = wmma4(a1, b, c1);
  }
}

// Hidden layer: LDS -> (WMMA, +bias, tanh) -> LDS. Padded cols get tanh(0)=0.
template <int KI, int NI, int KPAD, int NPAD>
static __device__ __forceinline__ void hidden_layer(
    const float* __restrict__ W, const float* __restrict__ B,
    const float* __restrict__ hin, float* __restrict__ hout, int lane) {
  const int half = lane >> 4;
  const int l16  = lane & 15;
#pragma unroll
  for (int nt = 0; nt < NPAD / 16; ++nt) {
    const int col = nt * 16 + l16;
    v8f c0, c1;
    tile_pair<KI, NI, KPAD>(W, hin, col, half, l16, c0, c1);
    const float bias = gload0(B, col < NI, col);
#pragma unroll
    for (int r = 0; r < 8; ++r) {
      const int m = r + half * 8;  // C layout: VGPR r -> row r + half*8
      hout[m * STRIDE + col]        = fast_tanh(c0[r] + bias);
      hout[(16 + m) * STRIDE + col] = fast_tanh(c1[r] + bias);
    }
  }
  lds_fence();
}

// Final layer: LDS -> registers; ACCUM=true adds into acc (second tower).
template <int KI, int NI, int KPAD, int NPAD, bool ACCUM>
static __device__ __forceinline__ void final_layer(
    const float* __restrict__ W, const float* __restrict__ B,
    const float* __restrict__ hin, v8f acc[2][NPAD / 16], int lane) {
  const int half = lane >> 4;
  const int l16  = lane & 15;
#pragma unroll
  for (int nt = 0; nt < NPAD / 16; ++nt) {
    const int col = nt * 16 + l16;
    v8f c0, c1;
    tile_pair<KI, NI, KPAD>(W, hin, col, half, l16, c0, c1);
    const float bias = gload0(B, col < NI, col);
#pragma unroll
    for (int r = 0; r < 8; ++r) {
      const float t0 = fast_tanh(c0[r] + bias);
      const float t1 = fast_tanh(c1[r] + bias);
      if (ACCUM) {
        acc[0][nt][r] += t0;
        acc[1][nt][r] += t1;
      } else {
        acc[0][nt][r] = t0;
        acc[1][nt][r] = t1;
      }
    }
  }
}

template <bool ACCUM>
static __device__ __forceinline__ void run_tower(
    const float* __restrict__ x, long m0, const float* W1, const float* B1,
    const float* W2, const float* B2, const float* W3, const float* B3,
    const float* W4, const float* B4, float* __restrict__ buf0,
    float* __restrict__ buf1, v8f acc[2][7], int lane) {
  // Stage x (dim 2 -> K padded to 4): all 32 lanes, one sample row each.
  const v2f xv = *(const v2f*)(x + (m0 + lane) * 2);  // 8B-aligned b64 load
  buf0[lane * STRIDE + 0] = xv.x;
  buf0[lane * STRIDE + 1] = xv.y;
  buf0[lane * STRIDE + 2] = 0.f;
  buf0[lane * STRIDE + 3] = 0.f;
  lds_fence();
  hidden_layer<2, 10, 4, 16>(W1, B1, buf0, buf1, lane);
  hidden_layer<10, 20, 16, 32>(W2, B2, buf1, buf0, lane);
  hidden_layer<20, 50, 32, 64>(W3, B3, buf0, buf1, lane);
  final_layer<50, 100, 64, 112, ACCUM>(W4, B4, buf1, acc, lane);
}

__global__ __launch_bounds__(WAVES_PER_BLOCK * LANES) void
double_decoder_kernel(const float* __restrict__ x,
                      const float* muW1, const float* mub1, const float* muW2,
                      const float* mub2, const float* muW3, const float* mub3,
                      const float* muW4, const float* mub4, const float* sgW1,
                      const float* sgb1, const float* sgW2, const float* sgb2,
                      const float* sgW3, const float* sgb3, const float* sgW4,
                      const float* sgb4, float* __restrict__ out) {
  __shared__ float lds[WAVES_PER_BLOCK * 2 * M_PER_WAVE * STRIDE];  // ~56.5 KB
  const int lane = threadIdx.x & (LANES - 1);
  const int w    = threadIdx.x / LANES;
  const long m0  = ((long)blockIdx.x * WAVES_PER_BLOCK + w) * M_PER_WAVE;

  float* buf0 = &lds[(w * 2 + 0) * M_PER_WAVE * STRIDE];
  float* buf1 = &lds[(w * 2 + 1) * M_PER_WAVE * STRIDE];

  v8f acc[2][7];  // mu tower writes, sigma tower accumulates: acc = mu + sigma
  run_tower<false>(x, m0, muW1, mub1, muW2, mub2, muW3, mub3, muW4, mub4, buf0,
                   buf1, acc, lane);
  run_tower<true>(x, m0, sgW1, sgb1, sgW2, sgb2, sgW3, sgb3, sgW4, sgb4, buf0,
                  buf1, acc, lane);

  // out[(m0 + mt*16 + m)*100 + col], real columns only (100 of 112 padded).
  const int half = lane >> 4;
  const int l16  = lane & 15;
#pragma unroll
  for (int mt = 0; mt < 2; ++mt) {
#pragma unroll
    for (int nt = 0; nt < 7; ++nt) {
      const int col = nt * 16 + l16;
      if (col < 100) {
#pragma unroll
        for (int r = 0; r < 8; ++r) {
          const long m = m0 + mt * 16 + r + half * 8;
          out[m * 100 + col] = acc[mt][nt][r];
        }
      }
    }
  }
}

extern "C" void kernel_launch(void* const* d_in, const int* in_sizes, int n_in,
                              void* d_out, int out_size, void* d_ws,
                              size_t ws_size, hipStream_t stream) {
  (void)in_sizes; (void)n_in; (void)d_ws; (void)ws_size; (void)out_size;
  const float* x    = (const float*)d_in[0];
  const float* muW1 = (const float*)d_in[1];
  const float* mub1 = (const float*)d_in[2];
  const float* muW2 = (const float*)d_in[3];
  const float* mub2 = (const float*)d_in[4];
  const float* muW3 = (const float*)d_in[5];
  const float* mub3 = (const float*)d_in[6];
  const float* muW4 = (const float*)d_in[7];
  const float* mub4 = (const float*)d_in[8];
  const float* sgW1 = (const float*)d_in[9];
  const float* sgb1 = (const float*)d_in[10];
  const float* sgW2 = (const float*)d_in[11];
  const float* sgb2 = (const float*)d_in[12];
  const float* sgW3 = (const float*)d_in[13];
  const float* sgb3 = (const float*)d_in[14];
  const float* sgW4 = (const float*)d_in[15];
  const float* sgb4 = (const float*)d_in[16];
  float* out = (float*)d_out;

  const int n_samples = 262144;
  const int blocks = n_samples / (WAVES_PER_BLOCK * M_PER_WAVE);  // 4096
  double_decoder_kernel<<<blocks, WAVES_PER_BLOCK * LANES, 0, stream>>>(
      x, muW1, mub1, muW2, mub2, muW3, mub3, muW4, mub4, sgW1, sgb1, sgW2,
      sgb2, sgW3, sgb3, sgW4, sgb4, out);
}